// YOLOLoss_24635932410041
// MI455X (gfx1250) — compile-verified
//
#include <hip/hip_runtime.h>
#include <hip/hip_bf16.h>

typedef __attribute__((ext_vector_type(2))) float v2f;
typedef __attribute__((ext_vector_type(8))) float v8f;

#define NBATCH 16
#define NTGT   32
#define NSCALE 3
#define NANCH  3
#define NCLS   80
#define CH     (5 + NCLS)

// grid sizes per scale: 64, 32, 16
#define CELLS0 (NBATCH * NANCH * 64 * 64)  // 196608
#define CELLS1 (NBATCH * NANCH * 32 * 32)  //  49152
#define CELLS2 (NBATCH * NANCH * 16 * 16)  //  12288
#define TOTAL_CELLS (CELLS0 + CELLS1 + CELLS2)
#define GRID_WS_OFF 16   // floats; acc scalars live in ws[0..11]

__device__ __forceinline__ float softplus_f(float x) {
  // stable log(1+exp(x))
  return fmaxf(x, 0.0f) + log1pf(expf(-fabsf(x)));
}
// bce_logits(p,t) = t*sp(-p) + (1-t)*sp(p) == sp(p) - p*t  (exact identity)
__device__ __forceinline__ float bce_f(float p, float t) {
  return softplus_f(p) - p * t;
}

__device__ __forceinline__ float iou_xywh(float ax, float ay, float aw, float ah,
                                          float bx, float by, float bw, float bh) {
  float a0x = ax - 0.5f * aw, a0y = ay - 0.5f * ah;
  float a1x = ax + 0.5f * aw, a1y = ay + 0.5f * ah;
  float b0x = bx - 0.5f * bw, b0y = by - 0.5f * bh;
  float b1x = bx + 0.5f * bw, b1y = by + 0.5f * bh;
  float x0 = fmaxf(a0x, b0x), y0 = fmaxf(a0y, b0y);
  float x1 = fminf(a1x, b1x), y1 = fminf(a1y, b1y);
  float inter = (x1 - x0) * (y1 - y0) * (((x0 < x1) && (y0 < y1)) ? 1.0f : 0.0f);
  float areaA = (a1x - a0x) * (a1y - a0y);
  float areaB = (b1x - b0x) * (b1y - b0y);
  return inter / (areaA + areaB - inter);
}

// Exact f32 wave32 sum. One V_WMMA_F32_16X16X4_F32 with B==ones performs
// D[m][n] = v[m] + v[m+16]; each lane's 8 D VGPRs then hold 8 distinct rows,
// so 7 adds + one shfl_xor(16) completes the 32-lane sum (mult-by-1.0 is exact).
__device__ __forceinline__ float wave_sum(float v) {
#if __has_builtin(__builtin_amdgcn_wmma_f32_16x16x4_f32)
  v2f a; a.x = v;    a.y = 0.0f;   // A[m][k]: lanes 0-15 -> k=0/1, lanes 16-31 -> k=2/3
  v2f b; b.x = 1.0f; b.y = 1.0f;   // B = all ones (4x16)
  v8f c = {0.0f, 0.0f, 0.0f, 0.0f, 0.0f, 0.0f, 0.0f, 0.0f};
  v8f d = __builtin_amdgcn_wmma_f32_16x16x4_f32(
      /*neg_a=*/false, a, /*neg_b=*/false, b,
      /*c_mod=*/(short)0, c, /*reuse_a=*/false, /*reuse_b=*/false);
  float r = ((d[0] + d[1]) + (d[2] + d[3])) + ((d[4] + d[5]) + (d[6] + d[7]));
  r += __shfl_xor(r, 16, 32);      // combine rows 0-7 (lo half) with rows 8-15 (hi half)
  return r;
#else
  float r = v;
#pragma unroll
  for (int off = 16; off >= 1; off >>= 1) r += __shfl_xor(r, off, 32);
  return r;
#endif
}

// ---------------------------------------------------------------------------
// Kernel 1: per-(target, scale, anchor) gather work.
// acc[0..2]=cnt_s  acc[3..5]=sum(1-iou)_s  acc[6..8]=cls_bce_sum_s
// Also scatters t_obj = 1.0 into the per-scale grids (races benign: all write 1).
// ---------------------------------------------------------------------------
__global__ void yolo_targets_kernel(const float* __restrict__ inf0,
                                    const float* __restrict__ inf1,
                                    const float* __restrict__ inf2,
                                    const float* __restrict__ anchors,
                                    const float* __restrict__ strides,
                                    const float* __restrict__ targets,
                                    float* __restrict__ ws) {
  int tid = blockIdx.x * blockDim.x + threadIdx.x;
  const int total = NBATCH * NTGT * NSCALE * NANCH;   // 4608
  if (tid >= total) return;

  int t = tid / (NSCALE * NANCH);
  int rem = tid - t * (NSCALE * NANCH);
  int s = rem / NANCH;
  int a = rem - s * NANCH;
  int bimg = t / NTGT;

  const float* tg = targets + (size_t)t * 5;
  float stride = strides[s];
  float tbx = tg[0] / stride, tby = tg[1] / stride;
  float tbw = tg[2] / stride, tbh = tg[3] / stride;
  int cls_id = (int)tg[4];

  float cx = floorf(tbx), cy = floorf(tby);
  float fx = tbx - cx, fy = tby - cy;
  int ci = (int)cx, cj = (int)cy;

  float aw = anchors[(s * NANCH + a) * 2 + 0] / stride;
  float ah = anchors[(s * NANCH + a) * 2 + 1] / stride;

  // anchor-match mask: IoU(target frac box, anchor box at 0.5) > 0.5
  float miou = iou_xywh(fx, fy, tbw, tbh, 0.5f, 0.5f, aw, ah);
  if (!(miou > 0.5f)) return;

  const float* inf = (s == 0) ? inf0 : ((s == 1) ? inf1 : inf2);
  int g = 64 >> s;
  size_t cell = (((size_t)bimg * NANCH + a) * g + cj) * (size_t)g + ci;

  // scatter t_obj = 1 (equivalent to .at[...].max(1))
  size_t goff = (s == 0) ? 0 : ((s == 1) ? CELLS0 : (CELLS0 + CELLS1));
  ws[GRID_WS_OFF + goff + cell] = 1.0f;

  atomicAdd(&ws[0 + s], 1.0f);  // cnt

  const float* pr = inf + cell * CH;
  float px = 1.0f / (1.0f + expf(-pr[0]));
  float py = 1.0f / (1.0f + expf(-pr[1]));
  float pw = fminf(expf(pr[2]), 1000.0f) * aw;
  float ph = fminf(expf(pr[3]), 1000.0f) * ah;
  float biou = iou_xywh(px, py, pw, ph, fx, fy, tbw, tbh);
  atomicAdd(&ws[3 + s], 1.0f - biou);

  float csum = 0.0f;
#pragma unroll 4
  for (int c = 0; c < NCLS; ++c) {
    float p = pr[5 + c];
    float tt = (c == cls_id) ? 1.0f : 0.0f;
    csum += bce_f(p, tt);
  }
  atomicAdd(&ws[6 + s], csum);
}

// ---------------------------------------------------------------------------
// Kernel 2: objectness BCE over every cell of every scale (blockIdx.y = scale).
// Hierarchical reduction: WMMA wave sum -> LDS -> WMMA wave sum -> atomicAdd.
// acc[9..11] = obj_bce_sum_s
// ---------------------------------------------------------------------------
__global__ void yolo_obj_kernel(const float* __restrict__ inf0,
                                const float* __restrict__ inf1,
                                const float* __restrict__ inf2,
                                float* __restrict__ ws) {
  int s = blockIdx.y;
  const float* inf = (s == 0) ? inf0 : ((s == 1) ? inf1 : inf2);
  int ncells = (s == 0) ? CELLS0 : ((s == 1) ? CELLS1 : CELLS2);
  size_t goff = (s == 0) ? 0 : ((s == 1) ? CELLS0 : (CELLS0 + CELLS1));
  const float* tobj = ws + GRID_WS_OFF + goff;

  float local = 0.0f;
  int stride = gridDim.x * blockDim.x;
  for (int idx = blockIdx.x * blockDim.x + threadIdx.x; idx < ncells; idx += stride) {
    float p = inf[(size_t)idx * CH + 4];
    float t = tobj[idx];
    local += bce_f(p, t);
  }

  __shared__ float wsum[8];
  float r = wave_sum(local);
  int wid = threadIdx.x >> 5;
  int lane = threadIdx.x & 31;
  if (lane == 0) wsum[wid] = r;
  __syncthreads();
  if (wid == 0) {
    int nwaves = blockDim.x >> 5;
    float v = (lane < nwaves) ? wsum[lane] : 0.0f;
    float tot = wave_sum(v);
    if (lane == 0) atomicAdd(&ws[9 + s], tot);
  }
}

// ---------------------------------------------------------------------------
// Kernel 3: combine accumulators into the scalar loss.
// ---------------------------------------------------------------------------
__global__ void yolo_finalize_kernel(const float* __restrict__ ws,
                                     float* __restrict__ out) {
  if (threadIdx.x != 0 || blockIdx.x != 0) return;
  const float cells[3] = {(float)CELLS0, (float)CELLS1, (float)CELLS2};
  float box_loss = 0.0f, obj_loss = 0.0f, cls_loss = 0.0f;
  for (int s = 0; s < NSCALE; ++s) {
    float cnt = fmaxf(ws[0 + s], 1.0f);
    box_loss += ws[3 + s] / cnt;
    cls_loss += ws[6 + s] / (cnt * (float)NCLS);
    obj_loss += ws[9 + s] / cells[s];
  }
  out[0] = box_loss * 3.54f + obj_loss * 64.3f + cls_loss * 37.4f;
}

extern "C" void kernel_launch(void* const* d_in, const int* in_sizes, int n_in,
                              void* d_out, int out_size, void* d_ws, size_t ws_size,
                              hipStream_t stream) {
  const float* inf0    = (const float*)d_in[0];
  const float* inf1    = (const float*)d_in[1];
  const float* inf2    = (const float*)d_in[2];
  const float* anchors = (const float*)d_in[3];
  const float* strides = (const float*)d_in[4];
  const float* targets = (const float*)d_in[5];
  float* ws  = (float*)d_ws;
  float* out = (float*)d_out;

  // zero accumulators + t_obj grids (capture-safe async memset)
  hipMemsetAsync(d_ws, 0, (size_t)(GRID_WS_OFF + TOTAL_CELLS) * sizeof(float), stream);

  // gather/scatter pass over (target, scale, anchor): 4608 threads
  {
    const int total = NBATCH * NTGT * NSCALE * NANCH;
    int threads = 256;
    int blocks = (total + threads - 1) / threads;
    yolo_targets_kernel<<<blocks, threads, 0, stream>>>(
        inf0, inf1, inf2, anchors, strides, targets, ws);
  }

  // objectness BCE reduction over all cells, one y-slice per scale
  yolo_obj_kernel<<<dim3(128, 3), dim3(256), 0, stream>>>(inf0, inf1, inf2, ws);

  yolo_finalize_kernel<<<1, 32, 0, stream>>>(ws, out);
}